// SingleQubitOperationLayer_3418793967680
// MI455X (gfx1250) — compile-verified
//
#include <hip/hip_runtime.h>
#include <math.h>

#define NQ 10
#define DIM 1024
#define PAD 33              // data / diag plane row stride (words)
#define BSP 34              // BS table row stride (words) -> 8B-aligned v2f rows
#define WAVES_PER_BLOCK 4
#define BLOCK_THREADS (32 * WAVES_PER_BLOCK)

typedef float v2f __attribute__((ext_vector_type(2)));
typedef float v8f __attribute__((ext_vector_type(8)));

// BS32 entries: bs32[m,k] = 2^-2.5 * i^popcount(m^k)
#define BSC 0.17677669529663687f

__device__ __forceinline__ float bs_re(int p) {
    return p == 0 ? BSC : (p == 2 ? -BSC : 0.0f);
}
__device__ __forceinline__ float bs_im(int p) {
    return p == 1 ? BSC : (p == 3 ? -BSC : 0.0f);
}

// ---------------------------------------------------------------------------
// Complex 32x32 matmul acc = BS32 * Z.  A-fragments loaded from LDS-resident
// BS tables (b64 per fragment); B-fragments from padded LDS data planes.
// WMMA f32 16x16x4: A 16x4 (2 VGPRs), B 4x16 (2 VGPRs), C/D 16x16 (8 VGPRs).
// ---------------------------------------------------------------------------
__device__ __forceinline__ void bs_matmul_lds(const float* Zr, const float* Zi,
                                              const float* bsR, const float* bsI,
                                              const float* bsNI,
                                              v8f (&accr)[2][2], v8f (&acci)[2][2],
                                              int lane) {
    const int half = lane >> 4;
    const int nn   = lane & 15;
    v8f zero = {};
#pragma unroll
    for (int ti = 0; ti < 2; ++ti)
#pragma unroll
        for (int tj = 0; tj < 2; ++tj) {
            accr[ti][tj] = zero;
            acci[ti][tj] = zero;
        }

#pragma unroll
    for (int kb = 0; kb < 8; ++kb) {
        const int krow = 4 * kb + 2 * half;   // lane's K base (vgpr v -> krow+v)
        v2f ar[2], ai[2], nai[2];
#pragma unroll
        for (int ti = 0; ti < 2; ++ti) {
            const int m = 16 * ti + nn;       // lane's A row
            ar[ti]  = *(const v2f*)(bsR  + m * BSP + krow);
            ai[ti]  = *(const v2f*)(bsI  + m * BSP + krow);
            nai[ti] = *(const v2f*)(bsNI + m * BSP + krow);
        }
#pragma unroll
        for (int tj = 0; tj < 2; ++tj) {
            const int col = 16 * tj + nn;
            v2f br, bi;
            br.x = Zr[krow * PAD + col];
            br.y = Zr[(krow + 1) * PAD + col];
            bi.x = Zi[krow * PAD + col];
            bi.y = Zi[(krow + 1) * PAD + col];
#pragma unroll
            for (int ti = 0; ti < 2; ++ti) {
                accr[ti][tj] = __builtin_amdgcn_wmma_f32_16x16x4_f32(
                    false, ar[ti],  false, br, (short)0, accr[ti][tj], false, false);
                accr[ti][tj] = __builtin_amdgcn_wmma_f32_16x16x4_f32(
                    false, nai[ti], false, bi, (short)0, accr[ti][tj], false, false);
                acci[ti][tj] = __builtin_amdgcn_wmma_f32_16x16x4_f32(
                    false, ar[ti],  false, bi, (short)0, acci[ti][tj], false, false);
                acci[ti][tj] = __builtin_amdgcn_wmma_f32_16x16x4_f32(
                    false, ai[ti],  false, br, (short)0, acci[ti][tj], false, false);
            }
        }
    }
}

// Fallback variant: A-fragments synthesized in VALU (no BS LDS tables needed).
__device__ __forceinline__ void bs_matmul_valu(const float* Zr, const float* Zi,
                                               v8f (&accr)[2][2], v8f (&acci)[2][2],
                                               int lane) {
    const int half = lane >> 4;
    const int nn   = lane & 15;
    v8f zero = {};
#pragma unroll
    for (int ti = 0; ti < 2; ++ti)
#pragma unroll
        for (int tj = 0; tj < 2; ++tj) {
            accr[ti][tj] = zero;
            acci[ti][tj] = zero;
        }
#pragma unroll
    for (int kb = 0; kb < 8; ++kb) {
        const int krow = 4 * kb + 2 * half;
        v2f ar[2], ai[2], nai[2];
#pragma unroll
        for (int ti = 0; ti < 2; ++ti) {
            const int m  = 16 * ti + nn;
            const int p0 = __popc(m ^ krow) & 3;
            const int p1 = __popc(m ^ (krow + 1)) & 3;
            ar[ti].x  = bs_re(p0);  ar[ti].y  = bs_re(p1);
            ai[ti].x  = bs_im(p0);  ai[ti].y  = bs_im(p1);
            nai[ti].x = -ai[ti].x;  nai[ti].y = -ai[ti].y;
        }
#pragma unroll
        for (int tj = 0; tj < 2; ++tj) {
            const int col = 16 * tj + nn;
            v2f br, bi;
            br.x = Zr[krow * PAD + col];
            br.y = Zr[(krow + 1) * PAD + col];
            bi.x = Zi[krow * PAD + col];
            bi.y = Zi[(krow + 1) * PAD + col];
#pragma unroll
            for (int ti = 0; ti < 2; ++ti) {
                accr[ti][tj] = __builtin_amdgcn_wmma_f32_16x16x4_f32(
                    false, ar[ti],  false, br, (short)0, accr[ti][tj], false, false);
                accr[ti][tj] = __builtin_amdgcn_wmma_f32_16x16x4_f32(
                    false, nai[ti], false, bi, (short)0, accr[ti][tj], false, false);
                acci[ti][tj] = __builtin_amdgcn_wmma_f32_16x16x4_f32(
                    false, ar[ti],  false, bi, (short)0, acci[ti][tj], false, false);
                acci[ti][tj] = __builtin_amdgcn_wmma_f32_16x16x4_f32(
                    false, ai[ti],  false, br, (short)0, acci[ti][tj], false, false);
            }
        }
    }
}

// ---------------------------------------------------------------------------
// Store C-layout accumulators TRANSPOSED into LDS; optionally multiply element
// U[m,n] by interleaved diag (re,im) at padded index of j = 32n + m first.
// C layout: lane L, vgpr v holds (m = 16*ti + v + 8*(L>>4), n = 16*tj + (L&15)).
// ---------------------------------------------------------------------------
template <bool WITH_DIAG>
__device__ __forceinline__ void store_acc_transposed(const v8f (&accr)[2][2],
                                                     const v8f (&acci)[2][2],
                                                     float* Zr, float* Zi,
                                                     const float* dRI, int lane) {
    const int half = lane >> 4;
    const int nn   = lane & 15;
#pragma unroll
    for (int ti = 0; ti < 2; ++ti)
#pragma unroll
        for (int tj = 0; tj < 2; ++tj)
#pragma unroll
            for (int v = 0; v < 8; ++v) {
                const int m   = 16 * ti + v + 8 * half;
                const int n   = 16 * tj + nn;
                const int idx = n * PAD + m;   // transposed slot == padded index of j=32n+m
                float r  = accr[ti][tj][v];
                float im = acci[ti][tj][v];
                if constexpr (WITH_DIAG) {
                    const v2f d  = *(const v2f*)(dRI + 2 * idx);
                    const float nr = r * d.x - im * d.y;
                    const float ni = r * d.y + im * d.x;
                    r = nr; im = ni;
                }
                Zr[idx] = r;
                Zi[idx] = im;
            }
}

// ---------------------------------------------------------------------------
// Kernel 1: compute the three diagonal vectors once into workspace.
// ws layout: [0,2048) d_in interleaved (re,im); [2048,4096) d_theta; [4096,6144) d_phi.
// ---------------------------------------------------------------------------
__global__ void diag_kernel(const float* __restrict__ alphas,
                            const float* __restrict__ betas,
                            const float* __restrict__ thetas,
                            const float* __restrict__ phis,
                            float* __restrict__ ws) {
    const int j = blockIdx.x * blockDim.x + threadIdx.x;
    if (j >= DIM) return;
    float pin = 0.f, pth = 0.f, pph = 0.f;
#pragma unroll
    for (int k = 0; k < NQ; ++k) {
        const int bit = (j >> (NQ - 1 - k)) & 1;
        pin += bit ? betas[k] : alphas[k];
        if (!bit) { pth += thetas[k]; pph += phis[k]; }
    }
    float s, c;
    sincosf(pin, &s, &c);
    ws[2 * j] = c;               ws[2 * j + 1] = s;
    sincosf(pth, &s, &c);
    ws[2 * DIM + 2 * j] = c;     ws[2 * DIM + 2 * j + 1] = s;
    sincosf(pph, &s, &c);
    ws[4 * DIM + 2 * j] = c;     ws[4 * DIM + 2 * j + 1] = s;
}

// ---------------------------------------------------------------------------
// Kernel 2 (main, workspace path): one wave per batch row.
// ---------------------------------------------------------------------------
__global__ __launch_bounds__(BLOCK_THREADS) void qphoton_main(
    const float* __restrict__ x, const float* __restrict__ ws,
    float* __restrict__ out, int batch) {
    __shared__ __align__(16) float buf[WAVES_PER_BLOCK][2][32 * PAD];
    __shared__ __align__(16) float dth[2 * 32 * PAD];   // interleaved (re,im), padded
    __shared__ __align__(16) float dph[2 * 32 * PAD];
    __shared__ __align__(16) float bsR[32 * BSP];
    __shared__ __align__(16) float bsI[32 * BSP];
    __shared__ __align__(16) float bsNI[32 * BSP];

    const int tid = threadIdx.x;

    // Copy theta/phi diagonals (interleaved) into padded LDS.
    for (int idx = tid; idx < 2 * DIM; idx += BLOCK_THREADS) {
        const int j = idx >> 1, c = idx & 1;
        const int p = ((j >> 5) * PAD + (j & 31)) * 2 + c;
        dth[p] = ws[2 * DIM + idx];
        dph[p] = ws[4 * DIM + idx];
    }
    // Build BS32 tables from popcount (no transcendentals).
    for (int idx = tid; idx < 1024; idx += BLOCK_THREADS) {
        const int m = idx >> 5, k = idx & 31;
        const int p = __popc(m ^ k) & 3;
        const float re = bs_re(p), im = bs_im(p);
        bsR[m * BSP + k]  = re;
        bsI[m * BSP + k]  = im;
        bsNI[m * BSP + k] = -im;
    }
    __syncthreads();

    const int lane = tid & 31;
    const int wv   = tid >> 5;
    const int b    = blockIdx.x * WAVES_PER_BLOCK + wv;
    if (b >= batch) return;   // wave-uniform; EXEC stays all-ones for WMMA

    float* Zr = buf[wv][0];
    float* Zi = buf[wv][1];

    const float* xr = x + (size_t)b * (2 * DIM);
    const float* xi = xr + DIM;

    // Load row (coalesced float4), apply d_in (from global ws, coalesced), to LDS.
#pragma unroll
    for (int it = 0; it < 8; ++it) {
        const int j0  = 4 * lane + 128 * it;
        const float4 vr = *(const float4*)(xr + j0);
        const float4 vi = *(const float4*)(xi + j0);
        const float4 d0 = *(const float4*)(ws + 2 * j0);
        const float4 d1 = *(const float4*)(ws + 2 * j0 + 4);
        const float drs[4] = {d0.x, d0.z, d1.x, d1.z};
        const float dis[4] = {d0.y, d0.w, d1.y, d1.w};
        const int idx = (j0 >> 5) * PAD + (j0 & 31);
#pragma unroll
        for (int e = 0; e < 4; ++e) {
            const float zr = (&vr.x)[e], zi = (&vi.x)[e];
            Zr[idx + e] = zr * drs[e] - zi * dis[e];
            Zi[idx + e] = zr * dis[e] + zi * drs[e];
        }
    }

    v8f accr[2][2], acci[2][2];

    // Stage 1: Y1 = BS32 * Z0 * BS32 (transpose trick), d_theta folded into store.
    bs_matmul_lds(Zr, Zi, bsR, bsI, bsNI, accr, acci, lane);
    store_acc_transposed<false>(accr, acci, Zr, Zi, nullptr, lane);
    bs_matmul_lds(Zr, Zi, bsR, bsI, bsNI, accr, acci, lane);
    store_acc_transposed<true>(accr, acci, Zr, Zi, dth, lane);
    // Stage 2: Y2 = BS32 * W * BS32, d_phi folded into final store.
    bs_matmul_lds(Zr, Zi, bsR, bsI, bsNI, accr, acci, lane);
    store_acc_transposed<false>(accr, acci, Zr, Zi, nullptr, lane);
    bs_matmul_lds(Zr, Zi, bsR, bsI, bsNI, accr, acci, lane);
    store_acc_transposed<true>(accr, acci, Zr, Zi, dph, lane);

    // Coalesced writeout.
    float* outr = out + (size_t)b * (2 * DIM);
    float* outi = outr + DIM;
#pragma unroll
    for (int it = 0; it < 8; ++it) {
        const int j0  = 4 * lane + 128 * it;
        const int idx = (j0 >> 5) * PAD + (j0 & 31);
        float4 vr, vi;
#pragma unroll
        for (int e = 0; e < 4; ++e) {
            (&vr.x)[e] = Zr[idx + e];
            (&vi.x)[e] = Zi[idx + e];
        }
        *(float4*)(outr + j0) = vr;
        *(float4*)(outi + j0) = vi;
    }
}

// ---------------------------------------------------------------------------
// Self-contained fallback (no workspace): diagonals built per block via sincos,
// BS fragments synthesized in VALU.
// ---------------------------------------------------------------------------
__global__ __launch_bounds__(BLOCK_THREADS) void qphoton_fallback(
    const float* __restrict__ x, const float* __restrict__ alphas,
    const float* __restrict__ betas, const float* __restrict__ thetas,
    const float* __restrict__ phis, float* __restrict__ out, int batch) {
    __shared__ __align__(16) float buf[WAVES_PER_BLOCK][2][32 * PAD];
    __shared__ __align__(16) float din[2 * 32 * PAD];   // interleaved, padded
    __shared__ __align__(16) float dth[2 * 32 * PAD];
    __shared__ __align__(16) float dph[2 * 32 * PAD];

    const int tid = threadIdx.x;
    for (int j = tid; j < DIM; j += BLOCK_THREADS) {
        float pin = 0.f, pth = 0.f, pph = 0.f;
#pragma unroll
        for (int k = 0; k < NQ; ++k) {
            const int bit = (j >> (NQ - 1 - k)) & 1;
            pin += bit ? betas[k] : alphas[k];
            if (!bit) { pth += thetas[k]; pph += phis[k]; }
        }
        const int p2 = ((j >> 5) * PAD + (j & 31)) * 2;
        float s, c;
        sincosf(pin, &s, &c);  din[p2] = c; din[p2 + 1] = s;
        sincosf(pth, &s, &c);  dth[p2] = c; dth[p2 + 1] = s;
        sincosf(pph, &s, &c);  dph[p2] = c; dph[p2 + 1] = s;
    }
    __syncthreads();

    const int lane = tid & 31;
    const int wv   = tid >> 5;
    const int b    = blockIdx.x * WAVES_PER_BLOCK + wv;
    if (b >= batch) return;

    float* Zr = buf[wv][0];
    float* Zi = buf[wv][1];

    const float* xr = x + (size_t)b * (2 * DIM);
    const float* xi = xr + DIM;

#pragma unroll
    for (int it = 0; it < 8; ++it) {
        const int j0  = 4 * lane + 128 * it;
        const float4 vr = *(const float4*)(xr + j0);
        const float4 vi = *(const float4*)(xi + j0);
        const int idx = (j0 >> 5) * PAD + (j0 & 31);
#pragma unroll
        for (int e = 0; e < 4; ++e) {
            const v2f d = *(const v2f*)(din + 2 * (idx + e));
            const float zr = (&vr.x)[e], zi = (&vi.x)[e];
            Zr[idx + e] = zr * d.x - zi * d.y;
            Zi[idx + e] = zr * d.y + zi * d.x;
        }
    }

    v8f accr[2][2], acci[2][2];
    bs_matmul_valu(Zr, Zi, accr, acci, lane);
    store_acc_transposed<false>(accr, acci, Zr, Zi, nullptr, lane);
    bs_matmul_valu(Zr, Zi, accr, acci, lane);
    store_acc_transposed<true>(accr, acci, Zr, Zi, dth, lane);
    bs_matmul_valu(Zr, Zi, accr, acci, lane);
    store_acc_transposed<false>(accr, acci, Zr, Zi, nullptr, lane);
    bs_matmul_valu(Zr, Zi, accr, acci, lane);
    store_acc_transposed<true>(accr, acci, Zr, Zi, dph, lane);

    float* outr = out + (size_t)b * (2 * DIM);
    float* outi = outr + DIM;
#pragma unroll
    for (int it = 0; it < 8; ++it) {
        const int j0  = 4 * lane + 128 * it;
        const int idx = (j0 >> 5) * PAD + (j0 & 31);
        float4 vr, vi;
#pragma unroll
        for (int e = 0; e < 4; ++e) {
            (&vr.x)[e] = Zr[idx + e];
            (&vi.x)[e] = Zi[idx + e];
        }
        *(float4*)(outr + j0) = vr;
        *(float4*)(outi + j0) = vi;
    }
}

extern "C" void kernel_launch(void* const* d_in, const int* in_sizes, int n_in,
                              void* d_out, int out_size, void* d_ws,
                              size_t ws_size, hipStream_t stream) {
    (void)n_in; (void)out_size;
    const float* x      = (const float*)d_in[0];
    const float* alphas = (const float*)d_in[1];
    const float* betas  = (const float*)d_in[2];
    const float* thetas = (const float*)d_in[3];
    const float* phis   = (const float*)d_in[4];
    float* out = (float*)d_out;

    const int batch = in_sizes[0] / (2 * DIM);
    const int grid  = (batch + WAVES_PER_BLOCK - 1) / WAVES_PER_BLOCK;

    if (ws_size >= (size_t)(6 * DIM) * sizeof(float) && d_ws != nullptr) {
        float* ws = (float*)d_ws;
        hipLaunchKernelGGL(diag_kernel, dim3((DIM + 127) / 128), dim3(128), 0,
                           stream, alphas, betas, thetas, phis, ws);
        hipLaunchKernelGGL(qphoton_main, dim3(grid), dim3(BLOCK_THREADS), 0,
                           stream, x, ws, out, batch);
    } else {
        hipLaunchKernelGGL(qphoton_fallback, dim3(grid), dim3(BLOCK_THREADS), 0,
                           stream, x, alphas, betas, thetas, phis, out, batch);
    }
}